// GCNII_30794915512599
// MI455X (gfx1250) — compile-verified
//
#include <hip/hip_runtime.h>
#include <hip/hip_bf16.h>
#include <math.h>

#define HIDDEN 128
#define ALPHA_C 0.1f

typedef float v2f __attribute__((ext_vector_type(2)));
typedef float v8f __attribute__((ext_vector_type(8)));

// ---------------------------------------------------------------------------
// fp32 WMMA 16x16 tile accumulator: D = A(16xK) * B(Kx16), K multiple of 4.
//   A 16x4 : lanes 0-15 -> M=lane, {v0=K0, v1=K1}; lanes 16-31 -> {v0=K2, v1=K3}
//   B 4x16 : lanes 0-15 -> N=lane, {v0=K0, v1=K1}; lanes 16-31 -> {v0=K2, v1=K3}
//   C/D    : VGPR r, lanes 0-15 -> M=r, N=lane; lanes 16-31 -> M=r+8, N=lane-16
// ---------------------------------------------------------------------------
__device__ __forceinline__ v8f wmma_tile_f32(const float* __restrict__ A, int lda,
                                             const float* __restrict__ B, int ldb,
                                             int K, int lane) {
  v8f acc = {};
  const int mn   = lane & 15;
  const int koff = (lane >> 4) << 1;   // 0 or 2
  for (int k = 0; k < K; k += 4) {
    v2f a, b;
    a.x = A[(size_t)mn * lda + (k + koff)];
    a.y = A[(size_t)mn * lda + (k + koff + 1)];
    b.x = B[(size_t)(k + koff)     * ldb + mn];
    b.y = B[(size_t)(k + koff + 1) * ldb + mn];
    acc = __builtin_amdgcn_wmma_f32_16x16x4_f32(false, a, false, b,
                                                (short)0, acc, false, false);
  }
  return acc;
}

// h = x @ w_in + b_in ; also h0 = h   (wave-per-tile, K=512)
__global__ void gemm_in_kernel(const float* __restrict__ X, const float* __restrict__ W,
                               const float* __restrict__ bias, float* __restrict__ H,
                               float* __restrict__ H0, int K, int ncols) {
  const int lane  = threadIdx.x & 31;
  const int tile  = (int)((blockIdx.x * blockDim.x + threadIdx.x) >> 5);
  const int tcols = ncols >> 4;
  const int trow  = tile / tcols, tcol = tile % tcols;
  v8f acc = wmma_tile_f32(X + (size_t)trow * 16 * K, K, W + tcol * 16, ncols, K, lane);
  const int col   = tcol * 16 + (lane & 15);
  const int rbase = trow * 16 + ((lane >> 4) << 3);
  const float bv  = bias[col];
  #pragma unroll
  for (int r = 0; r < 8; ++r) {
    float v = acc[r] + bv;
    size_t off = (size_t)(rbase + r) * ncols + col;
    H[off] = v;
    H0[off] = v;
  }
}

// h = elu((1-beta)*s + beta*(s @ W)); one block per 16-row stripe.
// The 16x128 A-tile is staged into LDS with async-to-LDS loads and shared by
// the block's 8 waves (one 16-col W tile each).
__global__ void gemm_layer_kernel(const float* __restrict__ S, const float* __restrict__ W,
                                  float* __restrict__ Hout, float beta) {
  __shared__ __align__(16) float Atile[16 * HIDDEN];   // 8 KB
  const int tid  = threadIdx.x;                        // 256 threads = 8 waves
  const int lane = tid & 31;
  const int wv   = tid >> 5;                           // wave id == tcol (0..7)
  const int trow = blockIdx.x;
  const float* Abase = S + (size_t)trow * 16 * HIDDEN;

  // async-stage A tile: 2048 floats = 512 x b128; 2 per thread
  #pragma unroll
  for (int i = 0; i < 2; ++i) {
    const int elt = (tid + i * 256) * 4;               // float index (16B granule)
    unsigned lds_addr = (unsigned)(size_t)(&Atile[elt]);
    unsigned long long gaddr = (unsigned long long)(size_t)(Abase + elt);
    asm volatile("global_load_async_to_lds_b128 %0, %1, off"
                 :: "v"(lds_addr), "v"(gaddr) : "memory");
  }
  asm volatile("s_wait_asynccnt 0" ::: "memory");
  __syncthreads();

  v8f acc = {};
  const int mn   = lane & 15;
  const int koff = (lane >> 4) << 1;
  const float* B = W + wv * 16;
  for (int k = 0; k < HIDDEN; k += 4) {
    v2f a, b;
    a.x = Atile[mn * HIDDEN + (k + koff)];
    a.y = Atile[mn * HIDDEN + (k + koff + 1)];
    b.x = B[(size_t)(k + koff)     * HIDDEN + mn];
    b.y = B[(size_t)(k + koff + 1) * HIDDEN + mn];
    acc = __builtin_amdgcn_wmma_f32_16x16x4_f32(false, a, false, b,
                                                (short)0, acc, false, false);
  }
  const int col  = wv * 16 + mn;
  const int rloc = (lane >> 4) << 3;                   // 0 or 8
  const int rbase = trow * 16 + rloc;
  #pragma unroll
  for (int r = 0; r < 8; ++r) {
    float sv = Atile[(rloc + r) * HIDDEN + col];       // s from LDS
    float v  = (1.0f - beta) * sv + beta * acc[r];
    Hout[(size_t)(rbase + r) * HIDDEN + col] = (v > 0.0f) ? v : (__expf(v) - 1.0f);
  }
}

// logits = h @ w_out + b_out
__global__ void gemm_out_kernel(const float* __restrict__ H, const float* __restrict__ W,
                                const float* __restrict__ bias, float* __restrict__ Out,
                                int K, int ncols) {
  const int lane  = threadIdx.x & 31;
  const int tile  = (int)((blockIdx.x * blockDim.x + threadIdx.x) >> 5);
  const int tcols = ncols >> 4;
  const int trow  = tile / tcols, tcol = tile % tcols;
  v8f acc = wmma_tile_f32(H + (size_t)trow * 16 * K, K, W + tcol * 16, ncols, K, lane);
  const int col   = tcol * 16 + (lane & 15);
  const int rbase = trow * 16 + ((lane >> 4) << 3);
  const float bv  = bias[col];
  #pragma unroll
  for (int r = 0; r < 8; ++r)
    Out[(size_t)(rbase + r) * ncols + col] = acc[r] + bv;
}

// ---------------------------------------------------------------------------
// Graph part
// ---------------------------------------------------------------------------
__global__ void degree_kernel(const long long* __restrict__ dst, float* __restrict__ deg, int E) {
  int e = blockIdx.x * blockDim.x + threadIdx.x;
  if (e < E) atomicAdd(&deg[(int)dst[e]], 1.0f);
}

__global__ void dinv_kernel(float* __restrict__ deg, int n) {
  int i = blockIdx.x * blockDim.x + threadIdx.x;
  if (i < n) deg[i] = rsqrtf(deg[i] + 1.0f);   // +1 self-loop; always > 0
}

// one-time: int64 -> int32 indices, per-edge norm = dinv[src]*dinv[dst]
__global__ void edge_prep_kernel(const long long* __restrict__ src64,
                                 const long long* __restrict__ dst64,
                                 const float* __restrict__ dinv,
                                 int* __restrict__ src32, int* __restrict__ dst32,
                                 float* __restrict__ norm, int E) {
  int e = blockIdx.x * blockDim.x + threadIdx.x;
  if (e >= E) return;
  int s = (int)src64[e], d = (int)dst64[e];
  src32[e] = s;
  dst32[e] = d;
  norm[e]  = dinv[s] * dinv[d];
}

// agg[dst] += norm * h[src] ; 32 threads/edge, 4 feats/thread (b128 gather)
__global__ void spmm_kernel(const int* __restrict__ src, const int* __restrict__ dst,
                            const float* __restrict__ norm, const float* __restrict__ h,
                            float* __restrict__ agg, int E) {
  long long idx = (long long)blockIdx.x * blockDim.x + threadIdx.x;
  int e  = (int)(idx >> 5);
  int f4 = (int)(idx & 31) << 2;
  if (e >= E) return;
  int s = src[e], d = dst[e];
  float w = norm[e];
  const float4 hv = *(const float4*)(h + (size_t)s * HIDDEN + f4);
  float* ap = agg + (size_t)d * HIDDEN + f4;
  atomicAdd(ap + 0, w * hv.x);
  atomicAdd(ap + 1, w * hv.y);
  atomicAdd(ap + 2, w * hv.z);
  atomicAdd(ap + 3, w * hv.w);
}

// s = (1-alpha)*(agg + dinv^2*h) + alpha*h0   (self-loop folded, float4)
__global__ void combine_kernel(const float4* __restrict__ agg, const float4* __restrict__ h,
                               const float4* __restrict__ h0, const float* __restrict__ dinv,
                               float4* __restrict__ s, int n4) {
  int idx = blockIdx.x * blockDim.x + threadIdx.x;
  if (idx >= n4) return;
  float dv = dinv[idx >> 5];        // HIDDEN/4 = 32 float4 per node
  float w  = dv * dv;
  float4 a = agg[idx], hv = h[idx], h0v = h0[idx];
  float4 o;
  o.x = (1.0f - ALPHA_C) * (a.x + w * hv.x) + ALPHA_C * h0v.x;
  o.y = (1.0f - ALPHA_C) * (a.y + w * hv.y) + ALPHA_C * h0v.y;
  o.z = (1.0f - ALPHA_C) * (a.z + w * hv.z) + ALPHA_C * h0v.z;
  o.w = (1.0f - ALPHA_C) * (a.w + w * hv.w) + ALPHA_C * h0v.w;
  s[idx] = o;
}

// in-place log_softmax over C=64 classes, one wave32 per row (2 classes/lane)
__global__ void log_softmax64_kernel(float* __restrict__ out, int nrows) {
  int wave = (int)((blockIdx.x * blockDim.x + threadIdx.x) >> 5);
  int lane = threadIdx.x & 31;
  if (wave >= nrows) return;
  float* row = out + (size_t)wave * 64;
  float a = row[lane], b = row[lane + 32];
  float m = fmaxf(a, b);
  #pragma unroll
  for (int o = 16; o > 0; o >>= 1) m = fmaxf(m, __shfl_xor(m, o, 32));
  float sum = __expf(a - m) + __expf(b - m);
  #pragma unroll
  for (int o = 16; o > 0; o >>= 1) sum += __shfl_xor(sum, o, 32);
  float lse = m + __logf(sum);
  row[lane]      = a - lse;
  row[lane + 32] = b - lse;
}

extern "C" void kernel_launch(void* const* d_in, const int* in_sizes, int n_in,
                              void* d_out, int out_size, void* d_ws, size_t ws_size,
                              hipStream_t stream) {
  const float*     x      = (const float*)d_in[0];
  const long long* ei     = (const long long*)d_in[1];  // int64 [2, E]
  const float*     w_in   = (const float*)d_in[2];
  const float*     b_in   = (const float*)d_in[3];
  const float*     conv_w = (const float*)d_in[4];
  const float*     w_out  = (const float*)d_in[5];
  const float*     b_out  = (const float*)d_in[6];

  const int F  = 512;
  const int Nn = in_sizes[0] / F;                 // 100000
  const int E  = in_sizes[1] / 2;                 // 1600000
  const int L  = in_sizes[4] / (HIDDEN * HIDDEN); // 8
  const int C  = in_sizes[5] / HIDDEN;            // 64

  const long long* src64 = ei;
  const long long* dst64 = ei + E;

  // workspace carve-up:
  // dinv[N] | h[N*H] | h0[N*H] | agg[N*H] | s[N*H] | norm[E] | src32[E] | dst32[E]
  float* dinv  = (float*)d_ws;
  float* h     = dinv + Nn;
  float* h0    = h    + (size_t)Nn * HIDDEN;
  float* agg   = h0   + (size_t)Nn * HIDDEN;
  float* sbuf  = agg  + (size_t)Nn * HIDDEN;
  float* norm  = sbuf + (size_t)Nn * HIDDEN;
  int*   src32 = (int*)(norm + E);
  int*   dst32 = src32 + E;

  // --- gcn_norm: deg -> dinv (in place), then per-edge norm + int32 indices ---
  hipMemsetAsync(dinv, 0, (size_t)Nn * sizeof(float), stream);
  degree_kernel<<<(E + 255) / 256, 256, 0, stream>>>(dst64, dinv, E);
  dinv_kernel<<<(Nn + 255) / 256, 256, 0, stream>>>(dinv, Nn);
  edge_prep_kernel<<<(E + 255) / 256, 256, 0, stream>>>(src64, dst64, dinv,
                                                        src32, dst32, norm, E);

  // --- h = x @ w_in + b_in ; h0 = h ---
  {
    int tiles = (Nn / 16) * (HIDDEN / 16);        // 50000, divisible by 8
    gemm_in_kernel<<<tiles / 8, 256, 0, stream>>>(x, w_in, b_in, h, h0, F, HIDDEN);
  }

  const long long spmm_threads = (long long)E * 32;
  const int spmm_blocks = (int)((spmm_threads + 255) / 256);
  const int nh  = Nn * HIDDEN;
  const int nh4 = nh / 4;

  for (int l = 0; l < L; ++l) {
    float beta = logf(0.5f / (float)(l + 1) + 1.0f);   // THETA = 0.5
    hipMemsetAsync(agg, 0, (size_t)nh * sizeof(float), stream);
    spmm_kernel<<<spmm_blocks, 256, 0, stream>>>(src32, dst32, norm, h, agg, E);
    combine_kernel<<<(nh4 + 255) / 256, 256, 0, stream>>>(
        (const float4*)agg, (const float4*)h, (const float4*)h0, dinv,
        (float4*)sbuf, nh4);
    gemm_layer_kernel<<<Nn / 16, 256, 0, stream>>>(
        sbuf, conv_w + (size_t)l * HIDDEN * HIDDEN, h, beta);
  }

  // --- logits = h @ w_out + b_out, then in-place log_softmax ---
  {
    int tiles = (Nn / 16) * (C / 16);             // 25000, divisible by 8
    gemm_out_kernel<<<tiles / 8, 256, 0, stream>>>(h, w_out, b_out, (float*)d_out,
                                                   HIDDEN, C);
    log_softmax64_kernel<<<(Nn + 7) / 8, 256, 0, stream>>>((float*)d_out, Nn);
  }
}